// Attn_74242804679156
// MI455X (gfx1250) — compile-verified
//
#include <hip/hip_runtime.h>
#include <hip/hip_bf16.h>

typedef __attribute__((ext_vector_type(2))) float v2f;
typedef __attribute__((ext_vector_type(8))) float v8f;

#define BB 64
#define SS 2048
#define HH 512
#define AA 3

// ---------------------------------------------------------------------------
// Kernel 1: ha[b,a] = sum_h hidden[b,h] * affect[h,a]   (tiny: 192 outputs)
// ---------------------------------------------------------------------------
__global__ void Attn_ha_kernel(const float* __restrict__ hidden,
                               const float* __restrict__ affect,
                               float* __restrict__ ha) {
    int t = threadIdx.x;
    if (t < BB * AA) {
        int b = t / AA, a = t % AA;
        const float* hrow = hidden + b * HH;
        float acc = 0.f;
        #pragma unroll 4
        for (int k = 0; k < HH; ++k) acc += hrow[k] * affect[k * AA + a];
        ha[t] = acc;
    }
}

// ---------------------------------------------------------------------------
// Kernel 2: energies via V_WMMA_F32_16X16X4_F32.
// One wave handles one batch b and one 16-row tile of s.
//   A (16x4 f32): lanes 0-15 -> rows M=0..15, K=0 (vgpr0), K=1 (vgpr1)
//                 lanes 16-31 -> rows M=0..15, K=2, K=3
//   -> each lane loads float2 enc[s0+(lane&15), b, k + 2*(lane>>4)]
//      (non-temporal: enc is a 256MB one-pass stream, larger than the 192MB L2)
//   B (4x16): h_b broadcast to all 16 columns -> same per-lane K pattern from
//      hidden (temporal: the 2KB row is re-read 128x per wave, keep it cached).
//   D: lane 0 holds M=0..7 (col 0), lane 16 holds M=8..15.
// ---------------------------------------------------------------------------
__global__ __launch_bounds__(256) void Attn_energy_kernel(
        const float* __restrict__ hidden,
        const float* __restrict__ enc,
        const float* __restrict__ emb,
        const float* __restrict__ ha,
        float* __restrict__ out) {
    const int lane = threadIdx.x & 31;
    const int wave = blockIdx.x * (blockDim.x >> 5) + (threadIdx.x >> 5);
    const int b    = wave >> 7;        // wave / (S/16)
    const int s0   = (wave & 127) << 4;

    const int m16 = lane & 15;           // A row handled by this lane
    const int khi = (lane >> 4) << 1;    // 0 for lanes 0-15, 2 for lanes 16-31

    const float* arow = enc    + (size_t)(s0 + m16) * (BB * HH) + (size_t)b * HH + khi;
    const float* brow = hidden + (size_t)b * HH + khi;

    v8f c = {};
    #pragma unroll 8
    for (int k = 0; k < HH; k += 4) {
        v2f af = __builtin_nontemporal_load((const v2f*)(arow + k)); // streaming
        v2f bf = *(const v2f*)(brow + k);                            // cached bcast
        c = __builtin_amdgcn_wmma_f32_16x16x4_f32(
                /*neg_a=*/false, af, /*neg_b=*/false, bf,
                /*c_mod=*/(short)0, c, /*reuse_a=*/false, /*reuse_b=*/false);
    }

    // Columns of D are identical; lane 0 owns rows 0..7, lane 16 owns rows 8..15.
    if ((lane & 15) == 0) {
        const int mbase = (lane >> 4) * 8;
        const float* h3 = ha + b * AA;
        const float h0 = h3[0], h1 = h3[1], h2 = h3[2];
        #pragma unroll
        for (int v = 0; v < 8; ++v) {
            const int s = s0 + mbase + v;
            const float* e3 = emb + (size_t)s * (BB * AA) + (size_t)b * AA;
            const float aff = h0 * e3[0] + h1 * e3[1] + h2 * e3[2];
            out[b * SS + s] = c[v] + aff;
        }
    }
}

// ---------------------------------------------------------------------------
// Kernel 3: in-place softmax over S per batch row. One block (8 waves) per b.
// Energies (512KB total) are L2-resident thanks to NT streaming of enc above.
// ---------------------------------------------------------------------------
__global__ __launch_bounds__(256) void Attn_softmax_kernel(float* __restrict__ out) {
    __shared__ float red[8];
    const int b = blockIdx.x;
    const int t = threadIdx.x;
    float* row = out + b * SS;

    float vals[8];
    float m = -__builtin_inff();
    #pragma unroll
    for (int i = 0; i < 8; ++i) {
        vals[i] = row[t + i * 256];
        m = fmaxf(m, vals[i]);
    }
    #pragma unroll
    for (int off = 16; off > 0; off >>= 1) m = fmaxf(m, __shfl_xor(m, off, 32));
    if ((t & 31) == 0) red[t >> 5] = m;
    __syncthreads();
    float bm = red[0];
    #pragma unroll
    for (int w = 1; w < 8; ++w) bm = fmaxf(bm, red[w]);

    float sum = 0.f;
    #pragma unroll
    for (int i = 0; i < 8; ++i) {
        vals[i] = __expf(vals[i] - bm);
        sum += vals[i];
    }
    #pragma unroll
    for (int off = 16; off > 0; off >>= 1) sum += __shfl_xor(sum, off, 32);
    __syncthreads();                 // done reading red for max
    if ((t & 31) == 0) red[t >> 5] = sum;
    __syncthreads();
    float bs = 0.f;
    #pragma unroll
    for (int w = 0; w < 8; ++w) bs += red[w];
    const float inv = 1.f / bs;
    #pragma unroll
    for (int i = 0; i < 8; ++i) row[t + i * 256] = vals[i] * inv;
}

// ---------------------------------------------------------------------------
extern "C" void kernel_launch(void* const* d_in, const int* in_sizes, int n_in,
                              void* d_out, int out_size, void* d_ws, size_t ws_size,
                              hipStream_t stream) {
    const float* hidden = (const float*)d_in[0];   // [1, B, H]
    const float* enc    = (const float*)d_in[1];   // [S, B, H]
    const float* emb    = (const float*)d_in[2];   // [S, B, 3]
    const float* affect = (const float*)d_in[3];   // [H, 3]
    float* out = (float*)d_out;                    // [B, 1, S]
    float* ha  = (float*)d_ws;                     // [B, 3] scratch (768 bytes)

    Attn_ha_kernel<<<1, 192, 0, stream>>>(hidden, affect, ha);

    const int total_waves = BB * (SS / 16);        // 8192
    Attn_energy_kernel<<<total_waves / 8, 256, 0, stream>>>(hidden, enc, emb, ha, out);

    Attn_softmax_kernel<<<BB, 256, 0, stream>>>(out);
}